// EvolveGCNWrapper_42803644072638
// MI455X (gfx1250) — compile-verified
//
#include <hip/hip_runtime.h>
#include <math.h>

// ---------------- problem constants (match reference) ----------------
#define NNODES 10000
#define NEDGES 320000
#define INDIM  128
#define HID    256

typedef float v2f __attribute__((ext_vector_type(2)));
typedef float v8f __attribute__((ext_vector_type(8)));

// =====================================================================
// WMMA f32 GEMM:  C[M,N] = A[M,K] * B (+ bias0 + bias1)
//   TRANSB=0: B is [K,N] row-major.
//   TRANSB=1: B is [N,K] row-major (i.e. C = A * B^T).
// One wave (32 threads) computes a 16x64 strip (1 M-tile x 4 N-tiles),
// reusing the A fragment across the 4 v_wmma_f32_16x16x4_f32 issues.
// grid = (N/64, M/16), block = 32 (wave32, EXEC all ones -> WMMA legal).
// =====================================================================
template <bool TRANSB>
__global__ __launch_bounds__(32) void wmma_gemm_f32(
    const float* __restrict__ A, const float* __restrict__ B,
    const float* __restrict__ bias0, const float* __restrict__ bias1,
    float* __restrict__ C, int M, int N, int K) {
  const int lane = threadIdx.x;   // 0..31
  const int half = lane >> 4;     // 0 or 1
  const int l    = lane & 15;
  const int m0   = blockIdx.y << 4;  // 16 rows
  const int n0   = blockIdx.x << 6;  // 64 cols = 4 tiles of 16
  (void)M;

  v8f acc0 = {}, acc1 = {}, acc2 = {}, acc3 = {};

  const float* arow = A + (size_t)(m0 + l) * K;

  for (int k0 = 0; k0 < K; k0 += 4) {
    const int ka = k0 + (half << 1);  // lanes 0-15 -> K=k0,k0+1 ; 16-31 -> k0+2,k0+3
    v2f a = *(const v2f*)(arow + ka);

    v2f b0, b1, b2, b3;
    if (TRANSB) {
      // B[n][k] contiguous in k -> float2 loads
      const float* bp0 = B + (size_t)(n0 + l) * K + ka;
      b0 = *(const v2f*)(bp0);
      b1 = *(const v2f*)(bp0 + (size_t)16 * K);
      b2 = *(const v2f*)(bp0 + (size_t)32 * K);
      b3 = *(const v2f*)(bp0 + (size_t)48 * K);
    } else {
      // B[k][n]: rows k and k+1 of each half
      const float* brow0 = B + (size_t)ka * N + n0 + l;
      const float* brow1 = brow0 + N;
      b0.x = brow0[0];  b0.y = brow1[0];
      b1.x = brow0[16]; b1.y = brow1[16];
      b2.x = brow0[32]; b2.y = brow1[32];
      b3.x = brow0[48]; b3.y = brow1[48];
    }

    acc0 = __builtin_amdgcn_wmma_f32_16x16x4_f32(false, a, false, b0, (short)0, acc0, false, false);
    acc1 = __builtin_amdgcn_wmma_f32_16x16x4_f32(false, a, false, b1, (short)0, acc1, false, false);
    acc2 = __builtin_amdgcn_wmma_f32_16x16x4_f32(false, a, false, b2, (short)0, acc2, false, false);
    acc3 = __builtin_amdgcn_wmma_f32_16x16x4_f32(false, a, false, b3, (short)0, acc3, false, false);
  }

  // Epilogue: VGPR r holds row m0+r (lanes 0-15) / m0+r+8 (lanes 16-31), col = n-tile base + l.
  const int rbase = m0 + (half << 3);
#pragma unroll
  for (int t = 0; t < 4; ++t) {
    const int col = n0 + t * 16 + l;
    float bsum = 0.0f;
    if (bias0) bsum += bias0[col];
    if (bias1) bsum += bias1[col];
    const v8f acc = (t == 0) ? acc0 : (t == 1) ? acc1 : (t == 2) ? acc2 : acc3;
#pragma unroll
    for (int r = 0; r < 8; ++r) {
      C[(size_t)(rbase + r) * N + col] = acc[r] + bsum;
    }
  }
}

// ---------------------------------------------------------------------
__global__ void elemwise_add_k(const float* __restrict__ a, const float* __restrict__ b,
                               float* __restrict__ c, int n) {
  int i = blockIdx.x * blockDim.x + threadIdx.x;
  if (i < n) c[i] = a[i] + b[i];
}

// gates: [HID, 4*HID] with torch gate order i,f,g,o ; iw: [HID, HID]
__global__ void lstm_cell_k(const float* __restrict__ gates, const float* __restrict__ iw,
                            float* __restrict__ W) {
  int idx = blockIdx.x * blockDim.x + threadIdx.x;
  if (idx >= HID * HID) return;
  int m = idx >> 8;
  int n = idx & (HID - 1);
  const float* g = gates + (size_t)m * 4 * HID;
  float ig = g[n];
  float fg = g[HID + n];
  float gg = g[2 * HID + n];
  float og = g[3 * HID + n];
  float si = 1.0f / (1.0f + expf(-ig));
  float sf = 1.0f / (1.0f + expf(-fg));
  float so = 1.0f / (1.0f + expf(-og));
  float c  = sf * iw[idx] + si * tanhf(gg);
  W[idx] = so * tanhf(c);
}

__global__ void deg_init_k(float* __restrict__ deg) {
  int i = blockIdx.x * blockDim.x + threadIdx.x;
  if (i < NNODES) deg[i] = 1.0f;  // self loop
}

__global__ void deg_count_k(const int* __restrict__ ei, float* __restrict__ deg) {
  int e = blockIdx.x * blockDim.x + threadIdx.x;
  if (e >= NEDGES) return;
  int d = ei[NEDGES + e];  // edge_index[1][e]
  atomicAdd(&deg[d], 1.0f);
}

__global__ void dinv_k(const float* __restrict__ deg, float* __restrict__ dinv) {
  int i = blockIdx.x * blockDim.x + threadIdx.x;
  if (i < NNODES) dinv[i] = rsqrtf(deg[i]);  // deg >= 1 always
}

// out = b_gcn + hw * dinv^2   (self-loop term folded in)
__global__ void out_init_k(const float* __restrict__ hw, const float* __restrict__ dinv,
                           const float* __restrict__ bg, float* __restrict__ out) {
  int idx = blockIdx.x * blockDim.x + threadIdx.x;
  if (idx >= NNODES * HID) return;
  int i = idx >> 8;
  int c = idx & (HID - 1);
  float di = dinv[i];
  out[idx] = bg[c] + hw[idx] * di * di;
}

// one 64-lane block per edge; each lane scatters 4 channels
__global__ __launch_bounds__(64) void scatter_edges_k(
    const int* __restrict__ ei, const float* __restrict__ hw,
    const float* __restrict__ dinv, float* __restrict__ out) {
  int e = blockIdx.x;
  int s = ei[e];           // edge_index[0][e]
  int d = ei[NEDGES + e];  // edge_index[1][e]
  float norm = dinv[s] * dinv[d];
  const float4* src4 = (const float4*)(hw + (size_t)s * HID);
  float* dst = out + (size_t)d * HID;
  int t = threadIdx.x;  // 0..63
  float4 v = src4[t];
  int c = t * 4;
  atomicAdd(dst + c + 0, v.x * norm);
  atomicAdd(dst + c + 1, v.y * norm);
  atomicAdd(dst + c + 2, v.z * norm);
  atomicAdd(dst + c + 3, v.w * norm);
}

// =====================================================================
extern "C" void kernel_launch(void* const* d_in, const int* in_sizes, int n_in,
                              void* d_out, int out_size, void* d_ws, size_t ws_size,
                              hipStream_t stream) {
  (void)in_sizes; (void)n_in; (void)out_size; (void)ws_size;

  const float* x     = (const float*)d_in[0];  // [N, 128]
  const int*   ei    = (const int*)  d_in[1];  // [2, E]
  const float* Wp    = (const float*)d_in[2];  // [128, 256]
  const float* bp    = (const float*)d_in[3];  // [256]
  const float* W_ih  = (const float*)d_in[4];  // [1024, 256]
  const float* W_hh  = (const float*)d_in[5];  // [1024, 256]
  const float* b_ih  = (const float*)d_in[6];  // [1024]
  const float* b_hh  = (const float*)d_in[7];  // [1024]
  const float* iw    = (const float*)d_in[8];  // [256, 256]
  const float* b_gcn = (const float*)d_in[9];  // [256]
  float* out = (float*)d_out;                  // [N, 256]

  // workspace carve-up (~23 MB of floats)
  float* ws    = (float*)d_ws;
  float* h     = ws;                                   // N*HID
  float* Wsum  = h    + (size_t)NNODES * HID;          // 4H*H
  float* gates = Wsum + (size_t)4 * HID * HID;         // H*4H
  float* Wevo  = gates + (size_t)4 * HID * HID;        // H*H
  float* hw    = Wevo + (size_t)HID * HID;             // N*HID
  float* deg   = hw   + (size_t)NNODES * HID;          // N
  float* dinv  = deg  + NNODES;                        // N

  // 1) h = x @ Wp + bp            (M=10000, K=128, N=256)
  {
    dim3 g(HID / 64, NNODES / 16);
    wmma_gemm_f32<false><<<g, 32, 0, stream>>>(x, Wp, bp, nullptr, h, NNODES, HID, INDIM);
  }

  // 2) weight evolution: gates = iw @ (W_ih + W_hh)^T + b_ih + b_hh
  {
    int tot = 4 * HID * HID;
    elemwise_add_k<<<(tot + 255) / 256, 256, 0, stream>>>(W_ih, W_hh, Wsum, tot);
    dim3 g((4 * HID) / 64, HID / 16);
    wmma_gemm_f32<true><<<g, 32, 0, stream>>>(iw, Wsum, b_ih, b_hh, gates, HID, 4 * HID, HID);
    lstm_cell_k<<<(HID * HID + 255) / 256, 256, 0, stream>>>(gates, iw, Wevo);
  }

  // 3) hw = h @ Wevo              (M=10000, K=256, N=256)
  {
    dim3 g(HID / 64, NNODES / 16);
    wmma_gemm_f32<false><<<g, 32, 0, stream>>>(h, Wevo, nullptr, nullptr, hw, NNODES, HID, HID);
  }

  // 4) symmetric-normalized propagation
  deg_init_k<<<(NNODES + 255) / 256, 256, 0, stream>>>(deg);
  deg_count_k<<<(NEDGES + 255) / 256, 256, 0, stream>>>(ei, deg);
  dinv_k<<<(NNODES + 255) / 256, 256, 0, stream>>>(deg, dinv);

  out_init_k<<<(NNODES * HID + 255) / 256, 256, 0, stream>>>(hw, dinv, b_gcn, out);
  scatter_edges_k<<<NEDGES, 64, 0, stream>>>(ei, hw, dinv, out);
}